// LFD_90486370993072
// MI455X (gfx1250) — compile-verified
//
#include <hip/hip_runtime.h>
#include <hip/hip_bf16.h>

#define TBL_STRIDE 260   // 260 % 64 == 4 -> bank = (4*row + col) & 63, rows spread across banks
#define CPAD 104         // padded row stride for cost / alignment matrices (even -> b64 loads)

typedef __attribute__((ext_vector_type(2))) float v2f;
typedef __attribute__((ext_vector_type(8))) float v8f;

__global__ void lfd_init(float* out, int n) {
    int i = blockIdx.x * blockDim.x + threadIdx.x;
    if (i < n) ((unsigned int*)out)[i] = 0x7F800000u;  // +inf bits
}

__global__ __launch_bounds__(256, 1)
void lfd_kernel(const float* __restrict__ q8,
                const int*   __restrict__ al,
                const int*   __restrict__ sA, const int* __restrict__ sF,
                const int*   __restrict__ sC, const int* __restrict__ sE,
                const int*   __restrict__ tA, const int* __restrict__ tF,
                const int*   __restrict__ tC, const int* __restrict__ tE,
                float* __restrict__ out, int M) {
    __shared__ float tbl[256 * TBL_STRIDE];  // padded fp32 q8_table (~260KB)
    __shared__ float costPad[16 * CPAD];     // cost[i][q*10+j], rows 10..15 zero
    __shared__ float Mlds[64 * CPAD];        // one-hot alignment matrix, rows 60..63 zero
    __shared__ float red[8];

    const int tid = threadIdx.x;
    const int bid = blockIdx.x;
    const int p = bid % 10;
    const int m = (bid / 10) % M;
    const int n = bid / (10 * M);

    // ---- stage q8_table into LDS with padded row stride (bank spreading) ----
    for (int f = tid; f < 16384; f += 256) {            // 16384 float4 = 65536 floats
        float4 v = ((const float4*)q8)[f];
        int r = f >> 6;                                  // table row
        int c = (f & 63) << 2;                           // table col (multiple of 4)
        *(float4*)&tbl[r * TBL_STRIDE + c] = v;          // 16B aligned stores
    }
    // ---- zero cost padding ----
    for (int z = tid; z < 16 * CPAD; z += 256) costPad[z] = 0.f;
    // ---- build one-hot alignment matrix Mlds[s][k] = (al[s, k%10] == k/10) ----
    for (int e = tid; e < 64 * 100; e += 256) {
        int s = e / 100, k = e % 100;
        int q = k / 10, j = k % 10;
        Mlds[s * CPAD + k] = (s < 60 && al[s * 20 + j] == q) ? 1.f : 0.f;
    }
    __syncthreads();

    // ---- gather phase: 1000 cost entries, 47 LDS gathers each ----
    for (int idx = tid; idx < 1000; idx += 256) {
        int i = idx / 100, k = idx % 100, q = k / 10, j = k % 10;
        const int sij = (n * 10 + i) * 10 + j;           // src (n,i,j)
        const int tpq = (m * 10 + p) * 10 + q;           // tgt (m,p,q)
        const int* sa = sA + sij * 35;
        const int* ta = tA + tpq * 35;
        float art = 0.f;
        #pragma unroll
        for (int a = 0; a < 35; ++a)
            art += tbl[sa[a] * TBL_STRIDE + ta[a]];
        const int* sf = sF + sij * 10;
        const int* tf = tF + tpq * 10;
        float fd = 0.f;
        #pragma unroll
        for (int b = 0; b < 10; ++b)
            fd += tbl[sf[b] * TBL_STRIDE + tf[b]];
        float cir = tbl[sC[sij] * TBL_STRIDE + tC[tpq]];
        float ecc = tbl[sE[sij] * TBL_STRIDE + tE[tpq]];
        costPad[i * CPAD + k] = art + 2.f * fd + 2.f * cir + ecc;
    }
    __syncthreads();

    // ---- WMMA phase: S[i,s] = sum_k cost[i,k] * M[s,k] ----
    // 4 waves; wave w owns s-tile [16w, 16w+16); 25 chained V_WMMA_F32_16X16X4_F32.
    // All operands are unconditional 8B-aligned LDS loads; EXEC stays all-ones.
    const int wave = tid >> 5, lane = tid & 31;
    if (wave < 4) {
        const int hf  = lane >> 4;       // lane half selects K pair {+0,+1} vs {+2,+3}
        const int l15 = lane & 15;
        const int sIdx = wave * 16 + l15;
        v8f acc = {0.f, 0.f, 0.f, 0.f, 0.f, 0.f, 0.f, 0.f};
        #pragma unroll
        for (int kk = 0; kk < 25; ++kk) {
            const int k0 = kk * 4 + hf * 2;
            // A fragment (16x4 f32): row = l15, K = k0 / k0+1
            v2f a = *(const v2f*)&costPad[l15 * CPAD + k0];
            // B fragment (4x16): N = l15 -> alignment s
            v2f b = *(const v2f*)&Mlds[sIdx * CPAD + k0];
            acc = __builtin_amdgcn_wmma_f32_16x16x4_f32(
                false, a, false, b, (short)0, acc, false, false);
        }
        // D layout: VGPR r, lanes 0-15 -> M=r, lanes 16-31 -> M=r+8; N = l15
        const bool sOK = sIdx < 60;
        float mn = __builtin_inff();
        #pragma unroll
        for (int r = 0; r < 8; ++r) {
            int Mr = r + 8 * hf;
            bool valid = sOK && (Mr < 10);
            mn = fminf(mn, valid ? acc[r] : __builtin_inff());
        }
        #pragma unroll
        for (int off = 16; off >= 1; off >>= 1)
            mn = fminf(mn, __shfl_xor(mn, off, 32));
        if (lane == 0) red[wave] = mn;
    }
    __syncthreads();
    if (tid == 0) {
        float v = fminf(fminf(red[0], red[1]), fminf(red[2], red[3]));
        // values are >= 0, so uint bit-compare == float compare
        atomicMin((unsigned int*)(out + n * M + m), __float_as_uint(v));
    }
}

extern "C" void kernel_launch(void* const* d_in, const int* in_sizes, int n_in,
                              void* d_out, int out_size, void* d_ws, size_t ws_size,
                              hipStream_t stream) {
    const float* q8 = (const float*)d_in[0];
    const int* al = (const int*)d_in[1];
    const int* sA = (const int*)d_in[2];
    const int* sF = (const int*)d_in[3];
    const int* sC = (const int*)d_in[4];
    const int* sE = (const int*)d_in[5];
    const int* tA = (const int*)d_in[6];
    const int* tF = (const int*)d_in[7];
    const int* tC = (const int*)d_in[8];
    const int* tE = (const int*)d_in[9];
    float* out = (float*)d_out;

    const int n = in_sizes[2] / 3500;   // src_ArtCoeff: (n,10,10,35)
    const int M = in_sizes[6] / 3500;   // tgt_ArtCoeff: (m,10,10,35)

    lfd_init<<<(out_size + 255) / 256, 256, 0, stream>>>(out, out_size);
    lfd_kernel<<<n * M * 10, 256, 0, stream>>>(q8, al, sA, sF, sC, sE,
                                               tA, tF, tC, tE, out, M);
}